// GlobalAttentionPool_55396488184387
// MI455X (gfx1250) — compile-verified
//
#include <hip/hip_runtime.h>

typedef __attribute__((ext_vector_type(2))) float v2f;
typedef __attribute__((ext_vector_type(8))) float v8f;

#define HID 64

// ---------- helpers: total-order int encoding of floats for atomicMax ----------
__device__ __forceinline__ int f2ord(float f) {
    int i = __float_as_int(f);
    return (i >= 0) ? i : (i ^ 0x7FFFFFFF);
}
__device__ __forceinline__ float ord2f(int i) {
    return __int_as_float((i >= 0) ? i : (i ^ 0x7FFFFFFF));
}

// ---------- init per-graph max (-inf as ordered int) and sum (0) ----------
__global__ void init_kernel(int* __restrict__ gmaxi, float* __restrict__ gsum, int g) {
    int i = blockIdx.x * blockDim.x + threadIdx.x;
    if (i < g) { gmaxi[i] = -2147483647 - 1; gsum[i] = 0.0f; }
}

// ---------- p[i] = x[i].W_rel ; score[i] = b_rel + x[i].W_root  (V_WMMA_F32_16X16X4_F32) ----------
// One wave per 16-node tile. A = 16x4 f32 tile of x; B = 4x16 with col0=W_rel, col1=W_root, rest 0.
// B is staged in LDS as [W_rel | W_root | zeros]; each lane gets one branch-free base pointer,
// so the inner loop is global_load_b64 (A) + ds_load_b64 (B) + v_wmma, no EXEC manipulation.
// A layout (32-bit A 16x4): lanes 0-15 M=0..15 hold {K=k,K=k+1}; lanes 16-31 hold {K=k+2,K=k+3}.
// D layout: c[r] holds (M = r + 8*(lane>=16), N = lane&15).
__global__ __launch_bounds__(256) void score_wmma_kernel(
    const float* __restrict__ x, const float* __restrict__ Wrel,
    const float* __restrict__ brel, const float* __restrict__ Wroot,
    float* __restrict__ p, float* __restrict__ score, int n)
{
    __shared__ float wlds[3 * HID];   // [0:64)=W_rel, [64:128)=W_root, [128:192)=0
    {
        const int t = threadIdx.x;
        if (t < HID)            wlds[t] = Wrel[t];
        else if (t < 2 * HID)   wlds[t] = Wroot[t - HID];
        else if (t < 3 * HID)   wlds[t] = 0.0f;
    }
    __syncthreads();                  // all waves reach this before any early-out

    const int lane = threadIdx.x & 31;
    const int wave = threadIdx.x >> 5;
    const int tile = blockIdx.x * 8 + wave;
    const int row0 = tile * 16;
    if (row0 >= n) return;            // wave-uniform exit, EXEC stays all-ones

    const int col  = lane & 15;       // N index (and M index for A rows)
    const int half = lane >> 4;       // K sub-pair select
    int arow = row0 + col;
    if (arow >= n) arow = n - 1;      // clamp; garbage rows discarded at store

    const float* __restrict__ xrow = x + (size_t)arow * HID + half * 2;
    const int wsel = (col < 2) ? col : 2;             // 0:W_rel 1:W_root 2:zeros
    const float* __restrict__ wrow = &wlds[wsel * HID + half * 2];

    v8f c = {0.f, 0.f, 0.f, 0.f, 0.f, 0.f, 0.f, 0.f};
#pragma unroll
    for (int k = 0; k < HID; k += 4) {
        v2f a = *(const v2f*)(xrow + k);              // global_load_b64, imm offset
        v2f b = *(const v2f*)(wrow + k);              // ds_load_b64, imm offset
        c = __builtin_amdgcn_wmma_f32_16x16x4_f32(false, a, false, b, (short)0, c,
                                                  false, false);
    }

    if (col < 2) {
        const float bias = (col == 1) ? *brel : 0.0f;
#pragma unroll
        for (int r = 0; r < 8; ++r) {
            int row = row0 + half * 8 + r;
            if (row < n) {
                float v = c[r] + bias;
                if (col == 0) p[row] = v;             // x . W_rel
                else          score[row] = v;         // b + x . W_root
            }
        }
    }
}

// ---------- scalar edge scatter: score[dst] += p[src] ----------
__global__ void edge_kernel(const int* __restrict__ src, const int* __restrict__ dst,
                            const float* __restrict__ p, float* __restrict__ score, int e)
{
    int i = blockIdx.x * blockDim.x + threadIdx.x;
    if (i < e) atomicAdd(&score[dst[i]], p[src[i]]);
}

// ---------- per-graph max of score ----------
__global__ void gmax_kernel(const float* __restrict__ score, const int* __restrict__ batch,
                            int* __restrict__ gmaxi, int n)
{
    int i = blockIdx.x * blockDim.x + threadIdx.x;
    if (i < n) atomicMax(&gmaxi[batch[i]], f2ord(score[i]));
}

// ---------- e[i] = exp(score - m[g]); gsum[g] += e[i]  (e stored over p) ----------
__global__ void expsum_kernel(const float* __restrict__ score, const int* __restrict__ batch,
                              const int* __restrict__ gmaxi, float* __restrict__ e,
                              float* __restrict__ gsum, int n)
{
    int i = blockIdx.x * blockDim.x + threadIdx.x;
    if (i < n) {
        int g = batch[i];
        float v = __expf(score[i] - ord2f(gmaxi[g]));
        e[i] = v;
        atomicAdd(&gsum[g], v);
    }
}

// ---------- segment boundaries via binary search (batch is sorted) ----------
__global__ void bounds_kernel(const int* __restrict__ batch, int* __restrict__ gstart,
                              int n, int g)
{
    int t = blockIdx.x * blockDim.x + threadIdx.x;
    if (t <= g) {
        int lo = 0, hi = n;
        while (lo < hi) { int mid = (lo + hi) >> 1; if (batch[mid] < t) lo = mid + 1; else hi = mid; }
        gstart[t] = lo;
    }
}

// ---------- pool: out[g][h] = sum_i x[i][h] * e[i]/gsum[g]  (one block per graph) ----------
__global__ __launch_bounds__(HID) void pool_kernel(
    const float* __restrict__ x, const float* __restrict__ e,
    const float* __restrict__ gsum, const int* __restrict__ gstart,
    float* __restrict__ out)
{
    int g = blockIdx.x;
    int h = threadIdx.x;
    int s = gstart[g], t = gstart[g + 1];
    float inv = 1.0f / gsum[g];       // empty segment -> loop skipped, acc stays 0
    float acc = 0.0f;
    for (int i = s; i < t; ++i)
        acc += x[(size_t)i * HID + h] * (e[i] * inv);
    out[(size_t)g * HID + h] = acc;
}

extern "C" void kernel_launch(void* const* d_in, const int* in_sizes, int n_in,
                              void* d_out, int out_size, void* d_ws, size_t ws_size,
                              hipStream_t stream)
{
    const float* x     = (const float*)d_in[0];
    const int*   ei    = (const int*)d_in[1];   // [2, E] flat: src then dst
    const int*   batch = (const int*)d_in[2];
    const float* Wrel  = (const float*)d_in[3];
    const float* brel  = (const float*)d_in[4];
    const float* Wroot = (const float*)d_in[5];

    const int N = in_sizes[0] / HID;
    const int E = in_sizes[1] / 2;
    const int G = out_size / HID;
    float* out = (float*)d_out;

    // workspace layout
    float* p      = (float*)d_ws;          // N  (later reused as e)
    float* score  = p + N;                 // N
    int*   gmaxi  = (int*)(score + N);     // G
    float* gsum   = (float*)(gmaxi + G);   // G
    int*   gstart = (int*)(gsum + G);      // G+1

    init_kernel<<<(G + 255) / 256, 256, 0, stream>>>(gmaxi, gsum, G);

    const int tiles = (N + 15) / 16;
    score_wmma_kernel<<<(tiles + 7) / 8, 256, 0, stream>>>(x, Wrel, brel, Wroot, p, score, N);

    edge_kernel<<<(E + 255) / 256, 256, 0, stream>>>(ei, ei + E, p, score, E);

    gmax_kernel<<<(N + 255) / 256, 256, 0, stream>>>(score, batch, gmaxi, N);

    expsum_kernel<<<(N + 255) / 256, 256, 0, stream>>>(score, batch, gmaxi, p, gsum, N);

    bounds_kernel<<<(G + 1 + 255) / 256, 256, 0, stream>>>(batch, gstart, N, G);

    pool_kernel<<<G, HID, 0, stream>>>(x, p, gsum, gstart, out);
}